// LSTM1_53060025975085
// MI455X (gfx1250) — compile-verified
//
#include <hip/hip_runtime.h>
#include <math.h>

// ---------------------------------------------------------------------------
// Time-decay LSTM for MI455X (gfx1250, wave32).
//  Phase 1: xg = x @ W_ih^T + b  via V_WMMA_F32_16X16X4_F32 (fp32 WMMA).
//  Phase 2: persistent-grid sequential scan; W_hh (67MB) streams from 192MB L2
//           (whole working set ~168MB stays L2-resident); decayed h double-
//           buffered in global; one monotonic atomic split-barrier per step.
//  Phase 3: attention pooling (dot, softmax, weighted sum).
// ---------------------------------------------------------------------------

typedef __attribute__((ext_vector_type(2))) float v2f;
typedef __attribute__((ext_vector_type(8))) float v8f;

#define T_LEN  2048
#define IN_DIM 1024
#define H_DIM  2048
#define G4H    8192

// workspace layout (in floats)
#define XG_OFF    0ull          // [T,4H]  16,777,216
#define GAMMA_OFF 16777216ull   // [T,H]    4,194,304
#define HS_OFF    20971520ull   // [T,H]    4,194,304
#define HDEC_OFF  25165824ull   // [2,H]        4,096 (double-buffered decayed h)
#define ATTN_OFF  25169920ull   // [T]          2,048
#define WSM_OFF   25171968ull   // [T]          2,048
#define CTR_OFF   25174016ull   // int barrier counter

#define NWG   128
#define UNITS 16                // H_DIM / NWG hidden units per workgroup

__device__ __forceinline__ float sigm(float z) { return 1.0f / (1.0f + expf(-z)); }

// ---------------------------------------------------------------------------
// Phase 1: xg[m][n] = sum_k x[m][k] * W_ih[n][k] + b_ih[n] + b_hh[n]
// One wave computes a 16x64 output strip (4 WMMA accumulators, A reused).
// ---------------------------------------------------------------------------
__global__ __launch_bounds__(256)
void xg_wmma_kernel(const float* __restrict__ x, const float* __restrict__ Wih,
                    const float* __restrict__ bih, const float* __restrict__ bhh,
                    float* __restrict__ xg) {
    const int lane = threadIdx.x & 31;
    const int wid  = blockIdx.x * 8 + (threadIdx.x >> 5);   // 16384 wave tasks
    const int mt = wid >> 7;          // 128 M-tiles of 16
    const int nt = wid & 127;         // 128 N-strips of 64
    const int m0 = mt * 16;
    const int n0 = nt * 64;

    const int half  = lane >> 4;      // 0: K pair {0,1}, 1: K pair {2,3}
    const int lcol  = lane & 15;
    const int am    = m0 + lcol;      // A-matrix row (M) for this lane

    v8f acc[4];
    acc[0] = (v8f)0.0f; acc[1] = (v8f)0.0f; acc[2] = (v8f)0.0f; acc[3] = (v8f)0.0f;

    const float* xrow = x + (size_t)am * IN_DIM;
    for (int kk = 0; kk < IN_DIM; kk += 4) {
        const int kb = kk + (half << 1);
        // A fragment (16x4 f32): lane holds A[am][kb], A[am][kb+1]
        v2f a = *(const v2f*)(xrow + kb);
#pragma unroll
        for (int u = 0; u < 4; ++u) {
            const int n = n0 + u * 16 + lcol;
            // B fragment (4x16 f32): lane holds B[kb][n], B[kb+1][n] = W_ih[n][kb..]
            v2f b = *(const v2f*)(Wih + (size_t)n * IN_DIM + kb);
            acc[u] = __builtin_amdgcn_wmma_f32_16x16x4_f32(
                false, a, false, b, (short)0, acc[u], false, false);
        }
    }

#pragma unroll
    for (int u = 0; u < 4; ++u) {
        const int n = n0 + u * 16 + lcol;
        const float bias = bih[n] + bhh[n];
#pragma unroll
        for (int r = 0; r < 8; ++r) {
            const int m = m0 + r + (half << 3);   // D layout: M = r + 8*(lane>>4)
            xg[(size_t)m * G4H + n] = acc[u][r] + bias;
        }
    }
}

// ---------------------------------------------------------------------------
// gamma[t][j] = (t==0) ? 1 : exp(-relu(dt_prev[t]*decay_w[j] + decay_b[j]))
// dt_prev[t] = time[t-1]-time[t-2] for t>=2 else 0
// ---------------------------------------------------------------------------
__global__ __launch_bounds__(256)
void gamma_kernel(const float* __restrict__ tm, const float* __restrict__ dw,
                  const float* __restrict__ db, float* __restrict__ gmm) {
    const int idx = blockIdx.x * 256 + threadIdx.x;     // [0, T*H)
    const int t = idx >> 11;
    const int j = idx & (H_DIM - 1);
    float g = 1.0f;
    if (t > 0) {
        const float dtp = (t >= 2) ? (tm[t - 1] - tm[t - 2]) : 0.0f;
        const float z = dtp * dw[j] + db[j];
        g = expf(-fmaxf(z, 0.0f));
    }
    gmm[idx] = g;
}

// ---------------------------------------------------------------------------
// init: hdec buffer0 = h0 (gamma at step 0 is 1), zero barrier counter
// ---------------------------------------------------------------------------
__global__ __launch_bounds__(256)
void init_kernel(const float* __restrict__ h0, float* __restrict__ hdec,
                 int* __restrict__ ctr) {
    const int j = blockIdx.x * 256 + threadIdx.x;
    if (j < H_DIM) hdec[j] = h0[j];
    if (j == 0) *ctr = 0;
}

// ---------------------------------------------------------------------------
// Phase 2: persistent scan. Each WG owns UNITS hidden units (all 4 gates).
// Per step: 4*UNITS GEMV dots of length H (4 threads/dot, W_hh hot in L2),
// gate nonlinearities, c/h update, write hs[t] and decayed h for step t+1.
// Device-wide split barrier: monotonic atomic counter, target (t+1)*NWG.
// Double-buffered hdec makes one barrier per step race-free (a WG can only
// be one step ahead of any other, and buffers alternate).
// ---------------------------------------------------------------------------
__global__ __launch_bounds__(256)
void scan_kernel(const float* __restrict__ Whh, const float* __restrict__ xg,
                 const float* __restrict__ gmm, const float* __restrict__ c0,
                 float* __restrict__ hs, float* __restrict__ hdec,
                 int* __restrict__ ctr) {
    __shared__ float c_s[UNITS];
    __shared__ float g_s[UNITS * 4];

    const int tid  = threadIdx.x;
    const int j0   = blockIdx.x * UNITS;
    const int dl   = tid >> 2;        // dot index 0..63 (= jl*4 + gate)
    const int p    = tid & 3;         // K-split quarter
    const int jl   = dl >> 2;         // local hidden unit 0..15
    const int gate = dl & 3;          // i,f,g,o
    const int row  = gate * H_DIM + j0 + jl;

    const float4* __restrict__ wrow =
        (const float4*)(Whh + (size_t)row * H_DIM + p * 512);

    if (tid < UNITS) c_s[tid] = c0[j0 + tid];
    __syncthreads();

    for (int t = 0; t < T_LEN; ++t) {
        const float4* __restrict__ hd =
            (const float4*)(hdec + (size_t)(t & 1) * H_DIM + p * 512);

        float s0 = 0.0f, s1 = 0.0f;
#pragma unroll 4
        for (int k = 0; k < 128; k += 2) {
            float4 w0 = wrow[k],     h0v = hd[k];
            float4 w1 = wrow[k + 1], h1v = hd[k + 1];
            s0 = fmaf(w0.x, h0v.x, fmaf(w0.y, h0v.y,
                 fmaf(w0.z, h0v.z, fmaf(w0.w, h0v.w, s0))));
            s1 = fmaf(w1.x, h1v.x, fmaf(w1.y, h1v.y,
                 fmaf(w1.z, h1v.z, fmaf(w1.w, h1v.w, s1))));
        }
        float sum = s0 + s1;
        sum += __shfl_xor(sum, 1);    // combine the 4 K-quarters (wave32)
        sum += __shfl_xor(sum, 2);

        if (p == 0)
            g_s[dl] = sum + xg[(size_t)t * G4H + gate * H_DIM + j0 + jl];
        __syncthreads();

        if (tid < UNITS) {
            const float i_ = sigm(g_s[tid * 4 + 0]);
            const float f_ = sigm(g_s[tid * 4 + 1]);
            const float gg = tanhf(g_s[tid * 4 + 2]);
            const float o_ = sigm(g_s[tid * 4 + 3]);
            const float c  = f_ * c_s[tid] + i_ * gg;
            c_s[tid] = c;
            const float h = o_ * tanhf(c);
            hs[(size_t)t * H_DIM + j0 + tid] = h;
            const float gn =
                (t + 1 < T_LEN) ? gmm[(size_t)(t + 1) * H_DIM + j0 + tid] : 1.0f;
            hdec[(size_t)((t + 1) & 1) * H_DIM + j0 + tid] = h * gn;
            __threadfence();          // release our stores to device scope
        }
        __syncthreads();

        if (tid == 0) {
            __hip_atomic_fetch_add(ctr, 1, __ATOMIC_RELEASE,
                                   __HIP_MEMORY_SCOPE_AGENT);
            const int target = (t + 1) * NWG;
            while (__hip_atomic_load(ctr, __ATOMIC_ACQUIRE,
                                     __HIP_MEMORY_SCOPE_AGENT) < target)
                __builtin_amdgcn_s_sleep(1);
        }
        __syncthreads();
        __threadfence();              // acquire: invalidate L0 before next reads
    }
}

// ---------------------------------------------------------------------------
// Phase 3a: attn[t] = dot(hs[t], hs[T-1])
// ---------------------------------------------------------------------------
__global__ __launch_bounds__(256)
void attn_kernel(const float* __restrict__ hs, float* __restrict__ attn) {
    __shared__ float red[256];
    const int t = blockIdx.x;
    const float* hrow = hs + (size_t)t * H_DIM;
    const float* hf   = hs + (size_t)(T_LEN - 1) * H_DIM;
    float s = 0.0f;
    for (int j = threadIdx.x; j < H_DIM; j += 256) s = fmaf(hrow[j], hf[j], s);
    red[threadIdx.x] = s;
    __syncthreads();
    for (int o = 128; o > 0; o >>= 1) {
        if (threadIdx.x < o) red[threadIdx.x] += red[threadIdx.x + o];
        __syncthreads();
    }
    if (threadIdx.x == 0) attn[t] = red[0];
}

// ---------------------------------------------------------------------------
// Phase 3b: w = softmax(attn) over T   (single workgroup)
// ---------------------------------------------------------------------------
__global__ __launch_bounds__(256)
void softmax_kernel(const float* __restrict__ attn, float* __restrict__ w) {
    __shared__ float red[256];
    float m = -INFINITY;
    for (int t = threadIdx.x; t < T_LEN; t += 256) m = fmaxf(m, attn[t]);
    red[threadIdx.x] = m;
    __syncthreads();
    for (int o = 128; o > 0; o >>= 1) {
        if (threadIdx.x < o) red[threadIdx.x] = fmaxf(red[threadIdx.x], red[threadIdx.x + o]);
        __syncthreads();
    }
    m = red[0];
    __syncthreads();
    float s = 0.0f;
    for (int t = threadIdx.x; t < T_LEN; t += 256) s += expf(attn[t] - m);
    red[threadIdx.x] = s;
    __syncthreads();
    for (int o = 128; o > 0; o >>= 1) {
        if (threadIdx.x < o) red[threadIdx.x] += red[threadIdx.x + o];
        __syncthreads();
    }
    const float inv = 1.0f / red[0];
    for (int t = threadIdx.x; t < T_LEN; t += 256)
        w[t] = expf(attn[t] - m) * inv;
}

// ---------------------------------------------------------------------------
// Phase 3c: context[j] = sum_t w[t] * hs[t][j]  (coalesced across j)
// ---------------------------------------------------------------------------
__global__ __launch_bounds__(256)
void context_kernel(const float* __restrict__ hs, const float* __restrict__ w,
                    float* __restrict__ out) {
    const int j = blockIdx.x * 256 + threadIdx.x;
    float acc = 0.0f;
    for (int t = 0; t < T_LEN; ++t)
        acc = fmaf(w[t], hs[(size_t)t * H_DIM + j], acc);
    out[j] = acc;
}

// ---------------------------------------------------------------------------
extern "C" void kernel_launch(void* const* d_in, const int* in_sizes, int n_in,
                              void* d_out, int out_size, void* d_ws, size_t ws_size,
                              hipStream_t stream) {
    const float* x    = (const float*)d_in[0];
    const float* tm   = (const float*)d_in[1];
    const float* Wih  = (const float*)d_in[2];
    const float* Whh  = (const float*)d_in[3];
    const float* bih  = (const float*)d_in[4];
    const float* bhh  = (const float*)d_in[5];
    const float* dw   = (const float*)d_in[6];
    const float* db   = (const float*)d_in[7];
    const float* h0   = (const float*)d_in[8];
    const float* c0   = (const float*)d_in[9];
    float* out = (float*)d_out;

    float* ws    = (float*)d_ws;
    float* xg    = ws + XG_OFF;
    float* gmm   = ws + GAMMA_OFF;
    float* hs    = ws + HS_OFF;
    float* hdec  = ws + HDEC_OFF;
    float* attn  = ws + ATTN_OFF;
    float* wsm   = ws + WSM_OFF;
    int*   ctr   = (int*)(ws + CTR_OFF);

    xg_wmma_kernel<<<2048, 256, 0, stream>>>(x, Wih, bih, bhh, xg);
    gamma_kernel<<<(T_LEN * H_DIM) / 256, 256, 0, stream>>>(tm, dw, db, gmm);
    init_kernel<<<(H_DIM + 255) / 256, 256, 0, stream>>>(h0, hdec, ctr);
    scan_kernel<<<NWG, 256, 0, stream>>>(Whh, xg, gmm, c0, hs, hdec, ctr);
    attn_kernel<<<T_LEN, 256, 0, stream>>>(hs, attn);
    softmax_kernel<<<1, 256, 0, stream>>>(attn, wsm);
    context_kernel<<<H_DIM / 256, 256, 0, stream>>>(hs, wsm, out);
}